// SelfAttn_55456617726045
// MI455X (gfx1250) — compile-verified
//
#include <hip/hip_runtime.h>
#include <hip/hip_bf16.h>

typedef __attribute__((ext_vector_type(2))) float v2f;
typedef __attribute__((ext_vector_type(8))) float v8f;

#define NPIX 4096   // 64*64
#define NB   4
#define C64  64
#define KF   8
#define CH   32

// ---------------------------------------------------------------------------
// Fused-DPP 16-lane butterfly reductions (single VALU op per step, no
// canonicalize, no LDS).  Masks stay within each 16-lane half-wave:
// xor1 = quad_perm[1,0,3,2]; xor2 = quad_perm[2,3,0,1];
// xor4 ~ row_half_mirror; xor8 ~ row_mirror (valid once smaller groups
// are uniform, which the reduction order guarantees).
// ---------------------------------------------------------------------------
__device__ __forceinline__ float vmax(float a, float b) {
    float d;
    asm("v_max_num_f32 %0, %1, %2" : "=v"(d) : "v"(a), "v"(b));
    return d;
}
#define DPP_RED_OP(NAME, OP, CTRL)                                          \
    __device__ __forceinline__ float NAME(float v) {                        \
        float d;                                                            \
        asm(OP " %0, %1, %1 " CTRL                                          \
            " row_mask:0xf bank_mask:0xf bound_ctrl:1"                      \
            : "=v"(d) : "v"(v));                                            \
        return d;                                                           \
    }
DPP_RED_OP(dmax_x1, "v_max_num_f32_dpp", "quad_perm:[1,0,3,2]")
DPP_RED_OP(dmax_x2, "v_max_num_f32_dpp", "quad_perm:[2,3,0,1]")
DPP_RED_OP(dmax_x4, "v_max_num_f32_dpp", "row_half_mirror")
DPP_RED_OP(dmax_x8, "v_max_num_f32_dpp", "row_mirror")
DPP_RED_OP(dadd_x1, "v_add_f32_dpp",     "quad_perm:[1,0,3,2]")
DPP_RED_OP(dadd_x2, "v_add_f32_dpp",     "quad_perm:[2,3,0,1]")
DPP_RED_OP(dadd_x4, "v_add_f32_dpp",     "row_half_mirror")
DPP_RED_OP(dadd_x8, "v_add_f32_dpp",     "row_mirror")

__device__ __forceinline__ float rowmax16(float v) {
    v = dmax_x1(v); v = dmax_x2(v); v = dmax_x4(v); v = dmax_x8(v);
    return v;
}
__device__ __forceinline__ float rowsum16(float v) {
    v = dadd_x1(v); v = dadd_x2(v); v = dadd_x4(v); v = dadd_x8(v);
    return v;
}

// ---------------------------------------------------------------------------
// Kernel 1: 1x1-conv projections.  f,g stored [b][K][N] (row-contig in N),
// h stored transposed hT[b][N][32] so PV B-fragments load 64B-contiguous rows.
// ---------------------------------------------------------------------------
__global__ __launch_bounds__(256) void proj_kernel(
    const float* __restrict__ x,
    const float* __restrict__ Wf, const float* __restrict__ bf,
    const float* __restrict__ Wg, const float* __restrict__ bg,
    const float* __restrict__ Wh, const float* __restrict__ bh,
    float* __restrict__ f, float* __restrict__ g, float* __restrict__ hT)
{
    __shared__ float sWf[KF * C64];
    __shared__ float sWg[KF * C64];
    __shared__ float sWh[CH * C64];
    __shared__ float sbf[KF], sbg[KF], sbh[CH];

    const int tid = threadIdx.x;
    for (int i = tid; i < KF * C64; i += 256) { sWf[i] = Wf[i]; sWg[i] = Wg[i]; }
    for (int i = tid; i < CH * C64; i += 256) { sWh[i] = Wh[i]; }
    if (tid < KF) { sbf[tid] = bf[tid]; sbg[tid] = bg[tid]; }
    if (tid < CH) { sbh[tid] = bh[tid]; }
    __syncthreads();

    const int b = blockIdx.x >> 4;                  // 16 blocks of 256 px per batch
    const int n = ((blockIdx.x & 15) << 8) + tid;   // pixel index

    float accF[KF], accG[KF], accH[CH];
#pragma unroll
    for (int k = 0; k < KF; ++k) { accF[k] = sbf[k]; accG[k] = sbg[k]; }
#pragma unroll
    for (int m = 0; m < CH; ++m) { accH[m] = sbh[m]; }

    const float* xb = x + (size_t)b * C64 * NPIX + n;
    for (int c = 0; c < C64; ++c) {
        const float xv = xb[c * NPIX];              // lane-contiguous in n
#pragma unroll
        for (int k = 0; k < KF; ++k) {
            accF[k] = fmaf(sWf[k * C64 + c], xv, accF[k]);
            accG[k] = fmaf(sWg[k * C64 + c], xv, accG[k]);
        }
#pragma unroll
        for (int m = 0; m < CH; ++m)
            accH[m] = fmaf(sWh[m * C64 + c], xv, accH[m]);
    }

    float* fb = f + (size_t)b * KF * NPIX;
    float* gb = g + (size_t)b * KF * NPIX;
#pragma unroll
    for (int k = 0; k < KF; ++k) {
        fb[k * NPIX + n] = accF[k];
        gb[k * NPIX + n] = accG[k];
    }
    float* hrow = hT + ((size_t)b * NPIX + n) * CH;
#pragma unroll
    for (int m = 0; m < CH; ++m) hrow[m] = accH[m];
}

// ---------------------------------------------------------------------------
// Kernel 2: flash attention.  One wave32 per 16-row i-tile; streams 128
// 32-wide j-tiles (two 16x16 S tiles per iteration -> softmax bookkeeping
// amortized over 2x the elements).  S = f_i^T g_j via chained
// V_WMMA_F32_16X16X4_F32 (K=8).  Scores are kept in the log2 domain
// (f pre-scaled by log2(e)) so every exponential is a single v_exp_f32.
// P staged through LDS only for the C-layout -> A-fragment reshape, then
// O += P * hT via 16 chained f32 WMMAs (K=32, two 16-col c-chunks).
// ---------------------------------------------------------------------------
__global__ __launch_bounds__(128) void attn_kernel(
    const float* __restrict__ f, const float* __restrict__ g,
    const float* __restrict__ hT, float* __restrict__ o_mid)
{
    __shared__ float plds[4][2 * 16 * 16];          // P staging, 2KB per wave

    const int wave = threadIdx.x >> 5;
    const int lane = threadIdx.x & 31;
    const int h    = lane >> 4;                     // half-wave select
    const int t    = lane & 15;

    const int tile   = blockIdx.x * 4 + wave;       // 0..1023
    const int b      = tile >> 8;                   // 256 i-tiles per batch
    const int i_base = (tile & 255) << 4;

    const float* fb = f  + (size_t)b * KF * NPIX;
    const float* gb = g  + (size_t)b * KF * NPIX;
    const float* hb = hT + (size_t)b * NPIX * CH;

    // A-fragments of f^T (16 x 8), loop-invariant, pre-scaled by log2(e)
    // so S comes out of the WMMA already in the log2 domain.
    // A layout (f32 16x4): lane t -> M=t ; VGPR0: K=2h, VGPR1: K=2h+1.
    const float LOG2E = 1.4426950408889634f;
    v2f af0, af1;
    af0.x = fb[(2 * h    ) * NPIX + i_base + t] * LOG2E;
    af0.y = fb[(2 * h + 1) * NPIX + i_base + t] * LOG2E;
    af1.x = fb[(4 + 2 * h) * NPIX + i_base + t] * LOG2E;
    af1.y = fb[(5 + 2 * h) * NPIX + i_base + t] * LOG2E;

    v8f O0 = {}, O1 = {};                           // 16 x 32 accumulator
    float m[8], l[8];
#pragma unroll
    for (int r = 0; r < 8; ++r) { m[r] = -1e30f; l[r] = 0.f; }

    float* pw = &plds[wave][0];

    for (int jt = 0; jt < 128; ++jt) {
        const int j0 = jt << 5;                     // tile0 at j0, tile1 at j0+16

        // B-fragments of g (8 x 16) for both j sub-tiles.
        v2f bgA0, bgA1, bgB0, bgB1;
        bgA0.x = gb[(2 * h    ) * NPIX + j0 + t];
        bgA0.y = gb[(2 * h + 1) * NPIX + j0 + t];
        bgA1.x = gb[(4 + 2 * h) * NPIX + j0 + t];
        bgA1.y = gb[(5 + 2 * h) * NPIX + j0 + t];
        bgB0.x = gb[(2 * h    ) * NPIX + j0 + 16 + t];
        bgB0.y = gb[(2 * h + 1) * NPIX + j0 + 16 + t];
        bgB1.x = gb[(4 + 2 * h) * NPIX + j0 + 16 + t];
        bgB1.y = gb[(5 + 2 * h) * NPIX + j0 + 16 + t];

        v8f S0 = {}, S1 = {};
        S0 = __builtin_amdgcn_wmma_f32_16x16x4_f32(false, af0, false, bgA0,
                                                   (short)0, S0, false, false);
        S0 = __builtin_amdgcn_wmma_f32_16x16x4_f32(false, af1, false, bgA1,
                                                   (short)0, S0, false, false);
        S1 = __builtin_amdgcn_wmma_f32_16x16x4_f32(false, af0, false, bgB0,
                                                   (short)0, S1, false, false);
        S1 = __builtin_amdgcn_wmma_f32_16x16x4_f32(false, af1, false, bgB1,
                                                   (short)0, S1, false, false);

        // --- online softmax (log2 domain); row M = r + 8h spans this half ---
#pragma unroll
        for (int r = 0; r < 8; ++r) {
            const float mx = rowmax16(vmax(S0[r], S1[r]));
            const float mn = vmax(m[r], mx);
            const float scale = __builtin_amdgcn_exp2f(m[r] - mn);
            const float p0 = __builtin_amdgcn_exp2f(S0[r] - mn);
            const float p1 = __builtin_amdgcn_exp2f(S1[r] - mn);
            const float su = rowsum16(p0 + p1);
            l[r] = fmaf(l[r], scale, su);
            m[r] = mn;
            O0[r] *= scale;
            O1[r] *= scale;
            S0[r] = p0;
            S1[r] = p1;
        }

        // --- stage both P tiles (C-layout) into LDS, reload as A-fragments ---
#pragma unroll
        for (int r = 0; r < 8; ++r) {
            pw[(r + 8 * h) * 16 + t]       = S0[r];   // P0[M][N]
            pw[256 + (r + 8 * h) * 16 + t] = S1[r];   // P1[M][N]
        }
        asm volatile("s_wait_dscnt 0" ::: "memory"); // wave-private RAW on LDS

        v2f ap0[4], ap1[4];
#pragma unroll
        for (int s4 = 0; s4 < 4; ++s4) {             // K-slice base = 4*s4
            ap0[s4].x = pw[t * 16 + s4 * 4 + 2 * h];
            ap0[s4].y = pw[t * 16 + s4 * 4 + 2 * h + 1];
            ap1[s4].x = pw[256 + t * 16 + s4 * 4 + 2 * h];
            ap1[s4].y = pw[256 + t * 16 + s4 * 4 + 2 * h + 1];
        }
        asm volatile("s_wait_dscnt 0" ::: "memory"); // loads done before overwrite

        // --- O(16x32) += P(16x32) * H(32x32), K=32 as 8 f32-WMMA slices ---
#pragma unroll
        for (int s4 = 0; s4 < 4; ++s4) {
            const int k0 = s4 * 4;
            const float* hr0 = hb + (size_t)(j0 + k0 + 2 * h    ) * CH;
            const float* hr1 = hb + (size_t)(j0 + k0 + 2 * h + 1) * CH;
            v2f bh0, bh1;
            bh0.x = hr0[t];        bh0.y = hr1[t];        // c-chunk 0..15
            bh1.x = hr0[16 + t];   bh1.y = hr1[16 + t];   // c-chunk 16..31
            O0 = __builtin_amdgcn_wmma_f32_16x16x4_f32(false, ap0[s4], false, bh0,
                                                       (short)0, O0, false, false);
            O1 = __builtin_amdgcn_wmma_f32_16x16x4_f32(false, ap0[s4], false, bh1,
                                                       (short)0, O1, false, false);
        }
#pragma unroll
        for (int s4 = 0; s4 < 4; ++s4) {
            const int k0 = s4 * 4;
            const float* hr0 = hb + (size_t)(j0 + 16 + k0 + 2 * h    ) * CH;
            const float* hr1 = hb + (size_t)(j0 + 16 + k0 + 2 * h + 1) * CH;
            v2f bh0, bh1;
            bh0.x = hr0[t];        bh0.y = hr1[t];
            bh1.x = hr0[16 + t];   bh1.y = hr1[16 + t];
            O0 = __builtin_amdgcn_wmma_f32_16x16x4_f32(false, ap1[s4], false, bh0,
                                                       (short)0, O0, false, false);
            O1 = __builtin_amdgcn_wmma_f32_16x16x4_f32(false, ap1[s4], false, bh1,
                                                       (short)0, O1, false, false);
        }
    }

    // --- normalize by row sums and store o_mid[b][N][32] ---
#pragma unroll
    for (int r = 0; r < 8; ++r) {
        const float inv = 1.0f / l[r];
        const int i = i_base + r + 8 * h;            // C/D layout: M = r + 8h
        float* orow = o_mid + ((size_t)b * NPIX + i) * CH;
        orow[t]      = O0[r] * inv;
        orow[16 + t] = O1[r] * inv;
    }
}

// ---------------------------------------------------------------------------
// Kernel 3: out = gamma * (Wv @ o_mid + bv) + x
// ---------------------------------------------------------------------------
__global__ __launch_bounds__(256) void out_kernel(
    const float* __restrict__ x,
    const float* __restrict__ Wv, const float* __restrict__ bv,
    const float* __restrict__ gamma,
    const float* __restrict__ o_mid, float* __restrict__ out)
{
    __shared__ float sWv[C64 * CH];
    __shared__ float sbv[C64];

    const int tid = threadIdx.x;
    for (int i = tid; i < C64 * CH; i += 256) sWv[i] = Wv[i];
    if (tid < C64) sbv[tid] = bv[tid];
    __syncthreads();

    const int b = blockIdx.x >> 4;
    const int n = ((blockIdx.x & 15) << 8) + tid;

    float om[CH];
    const float* orow = o_mid + ((size_t)b * NPIX + n) * CH;
#pragma unroll
    for (int ch = 0; ch < CH; ++ch) om[ch] = orow[ch];

    const float gm = gamma[0];
    for (int c = 0; c < C64; ++c) {
        float acc = sbv[c];
#pragma unroll
        for (int ch = 0; ch < CH; ++ch)
            acc = fmaf(sWv[c * CH + ch], om[ch], acc);
        const size_t idx = ((size_t)b * C64 + c) * NPIX + n;
        out[idx] = gm * acc + x[idx];
    }
}

// ---------------------------------------------------------------------------
extern "C" void kernel_launch(void* const* d_in, const int* in_sizes, int n_in,
                              void* d_out, int out_size, void* d_ws, size_t ws_size,
                              hipStream_t stream) {
    const float* x     = (const float*)d_in[0];
    const float* Wf    = (const float*)d_in[1];
    const float* bf    = (const float*)d_in[2];
    const float* Wg    = (const float*)d_in[3];
    const float* bg    = (const float*)d_in[4];
    const float* Wh    = (const float*)d_in[5];
    const float* bh    = (const float*)d_in[6];
    const float* Wv    = (const float*)d_in[7];
    const float* bv    = (const float*)d_in[8];
    const float* gamma = (const float*)d_in[9];
    float* out = (float*)d_out;

    // workspace carve-up (5 MB total, fp32)
    float* f     = (float*)d_ws;                        // 4*8*4096
    float* g     = f   + (size_t)NB * KF * NPIX;        // 4*8*4096
    float* hT    = g   + (size_t)NB * KF * NPIX;        // 4*4096*32
    float* o_mid = hT  + (size_t)NB * NPIX * CH;        // 4*4096*32

    proj_kernel<<<NB * 16, 256, 0, stream>>>(x, Wf, bf, Wg, bg, Wh, bh, f, g, hT);
    attn_kernel<<<(NB * (NPIX / 16)) / 4, 128, 0, stream>>>(f, g, hT, o_mid);
    out_kernel<<<NB * 16, 256, 0, stream>>>(x, Wv, bv, gamma, o_mid, out);
}